// Pixelwise_70626442215665
// MI455X (gfx1250) — compile-verified
//
#include <hip/hip_runtime.h>
#include <math.h>

typedef __attribute__((ext_vector_type(2))) float v2f;
typedef __attribute__((ext_vector_type(8))) float v8f;

#define N_SAMP   10000
#define N_TILES  625          // 10000 / 16 candidate tiles
#define N_PIX    16384        // 1*128*128
#define CHUNK    2500         // ModFs LDS staging chunk (30 KB)

static constexpr double SOL_MM   = 299792458.0 * 1000.0;          // mm/s
static constexpr double F_MAXd   = SOL_MM / (2.0 * 10000.0);
static constexpr double TAU_MINd = 1.0 / F_MAXd;
static constexpr double DTd      = TAU_MINd / 10000.0;
static constexpr double P_SRCd   = 1.0e9;
static constexpr double P_AMBd   = 1.0e6;
static constexpr double GAMMAd   = 1.0 / (1.0e-4 * 0.1 * (P_AMBd + P_SRCd));
static constexpr double ETOTALd  = TAU_MINd * P_SRCd;              // tau_min * pAveSource
static constexpr double BSCALEd  = GAMMAd * 1.0e-4 * (0.1 / TAU_MINd); // GAMMA*BETA*T_INT/TAU

// -------- kernel 1: DemodFs[n,k] = sum_o alpha[k,o]*sin(omega[k,o]*n + phi[k,o]) --------
__global__ void k_demod(const float* __restrict__ alpha, const float* __restrict__ omega,
                        const float* __restrict__ phi, float* __restrict__ demod) {
    int n = blockIdx.x * blockDim.x + threadIdx.x;
    if (n >= N_SAMP) return;
    float fn = (float)n;
#pragma unroll
    for (int k = 0; k < 3; ++k) {
        float s = 0.0f;
#pragma unroll
        for (int o = 0; o < 4; ++o)
            s += alpha[k * 4 + o] * sinf(omega[k * 4 + o] * fn + phi[k * 4 + o]);
        demod[n * 3 + k] = s;
    }
}

// -------- kernel 2: per-channel sums of ModFs and DemodFs (deterministic) --------
__global__ void k_sums(const float* __restrict__ mod, const float* __restrict__ demod,
                       float* __restrict__ sums) {
    __shared__ float red[256];
    float acc[6] = {0.f, 0.f, 0.f, 0.f, 0.f, 0.f};
    for (int n = threadIdx.x; n < N_SAMP; n += 256) {
        acc[0] += mod[n * 3 + 0];   acc[1] += mod[n * 3 + 1];   acc[2] += mod[n * 3 + 2];
        acc[3] += demod[n * 3 + 0]; acc[4] += demod[n * 3 + 1]; acc[5] += demod[n * 3 + 2];
    }
    for (int j = 0; j < 6; ++j) {
        red[threadIdx.x] = acc[j];
        __syncthreads();
        for (int s = 128; s > 0; s >>= 1) {
            if ((int)threadIdx.x < s) red[threadIdx.x] += red[threadIdx.x + s];
            __syncthreads();
        }
        if (threadIdx.x == 0) sums[j] = red[0];
        __syncthreads();
    }
}

// -------- kernel 3: CorrFs[n,k] = DT * sum_t ModFs_s[t,k]*Demod[(t+n)%N,k]
//          ModFs_s fold-in: DT*scale_k*acc = ETOTAL*acc/sumMod_k --------
__global__ void k_corr(const float* __restrict__ mod, const float* __restrict__ demod,
                       const float* __restrict__ sums, float* __restrict__ corr) {
    __shared__ float smod[CHUNK * 3];
    int n = blockIdx.x * blockDim.x + threadIdx.x;
    float a0 = 0.f, a1 = 0.f, a2 = 0.f;
    for (int c = 0; c < N_SAMP; c += CHUNK) {
        for (int i = threadIdx.x; i < CHUNK * 3; i += blockDim.x)
            smod[i] = mod[c * 3 + i];
        __syncthreads();
        if (n < N_SAMP) {
            int j = n + c; if (j >= N_SAMP) j -= N_SAMP;
            for (int t = 0; t < CHUNK; ++t) {
                float d0 = demod[j * 3 + 0], d1 = demod[j * 3 + 1], d2 = demod[j * 3 + 2];
                a0 = fmaf(smod[t * 3 + 0], d0, a0);
                a1 = fmaf(smod[t * 3 + 1], d1, a1);
                a2 = fmaf(smod[t * 3 + 2], d2, a2);
                ++j; if (j >= N_SAMP) j = 0;
            }
        }
        __syncthreads();
    }
    if (n < N_SAMP) {
        float et = (float)ETOTALd;
        corr[n * 3 + 0] = et * a0 / sums[0];
        corr[n * 3 + 1] = et * a1 / sums[1];
        corr[n * 3 + 2] = et * a2 / sums[2];
    }
}

// -------- kernel 4: z-normalize CorrFs (ddof=1), pack into WMMA B-fragment layout,
//          kappas[k] = sumDemod[k]*DT --------
__global__ void k_pack(const float* __restrict__ corr, const float* __restrict__ sums,
                       float* __restrict__ bp, float* __restrict__ consts) {
    __shared__ float red[256];
    __shared__ float meanv[3], invstd[3];
    float a3[3] = {0.f, 0.f, 0.f};
    for (int n = threadIdx.x; n < N_SAMP; n += 256) {
        a3[0] += corr[n * 3 + 0]; a3[1] += corr[n * 3 + 1]; a3[2] += corr[n * 3 + 2];
    }
    for (int k = 0; k < 3; ++k) {
        red[threadIdx.x] = a3[k];
        __syncthreads();
        for (int s = 128; s > 0; s >>= 1) {
            if ((int)threadIdx.x < s) red[threadIdx.x] += red[threadIdx.x + s];
            __syncthreads();
        }
        if (threadIdx.x == 0) meanv[k] = red[0] / (float)N_SAMP;
        __syncthreads();
    }
    a3[0] = a3[1] = a3[2] = 0.f;
    for (int n = threadIdx.x; n < N_SAMP; n += 256) {
        float d0 = corr[n * 3 + 0] - meanv[0];
        float d1 = corr[n * 3 + 1] - meanv[1];
        float d2 = corr[n * 3 + 2] - meanv[2];
        a3[0] = fmaf(d0, d0, a3[0]); a3[1] = fmaf(d1, d1, a3[1]); a3[2] = fmaf(d2, d2, a3[2]);
    }
    for (int k = 0; k < 3; ++k) {
        red[threadIdx.x] = a3[k];
        __syncthreads();
        for (int s = 128; s > 0; s >>= 1) {
            if ((int)threadIdx.x < s) red[threadIdx.x] += red[threadIdx.x + s];
            __syncthreads();
        }
        if (threadIdx.x == 0) invstd[k] = rsqrtf(red[0] / (float)(N_SAMP - 1));
        __syncthreads();
    }
    // B fragment per 16-candidate tile: float2 per lane; lanes 0-15 = (K0,K1), 16-31 = (K2,0)
    float2* p = (float2*)bp;
    for (int c = threadIdx.x; c < N_SAMP; c += 256) {
        int tile = c >> 4, col = c & 15;
        float n0 = (corr[c * 3 + 0] - meanv[0]) * invstd[0];
        float n1 = (corr[c * 3 + 1] - meanv[1]) * invstd[1];
        float n2 = (corr[c * 3 + 2] - meanv[2]) * invstd[2];
        p[tile * 32 + col]      = make_float2(n0, n1);
        p[tile * 32 + 16 + col] = make_float2(n2, 0.0f);
    }
    if (threadIdx.x < 3) consts[threadIdx.x] = sums[3 + threadIdx.x] * (float)DTd;
}

// -------- kernel 5: per-pixel ZNCC vs all candidates via V_WMMA_F32_16X16X4_F32 + argmax --
__global__ void __launch_bounds__(256)
k_decode(const float* __restrict__ gt, const float* __restrict__ corr,
         const float* __restrict__ bp, const float* __restrict__ consts,
         float* __restrict__ out) {
    const int lane = threadIdx.x & 31;
    const int wave = (blockIdx.x * blockDim.x + threadIdx.x) >> 5;  // pixel tile 0..1023
    const int m    = lane & 15;
    const bool hi  = lane >= 16;

    // Brightness + per-pixel z-norm over K=3 (ddof=1)
    const int   pix = wave * 16 + m;
    const float dep = gt[pix];
    int idx = (int)rintf(dep);
    idx = idx < 0 ? 0 : (idx > N_SAMP - 1 ? N_SAMP - 1 : idx);
    const float sc  = (float)BSCALEd, pa = (float)P_AMBd;
    const float b0 = sc * (corr[idx * 3 + 0] + pa * consts[0]);
    const float b1 = sc * (corr[idx * 3 + 1] + pa * consts[1]);
    const float b2 = sc * (corr[idx * 3 + 2] + pa * consts[2]);
    const float mu = (b0 + b1 + b2) * (1.0f / 3.0f);
    const float e0 = b0 - mu, e1 = b1 - mu, e2 = b2 - mu;
    const float inv = rsqrtf((e0 * e0 + e1 * e1 + e2 * e2) * 0.5f);  // ddof=1: /(K-1)

    // A fragment (16x4 f32): lanes 0-15 hold K=0/1, lanes 16-31 hold K=2/3(=0)
    v2f a;
    a.x = hi ? e2 * inv : e0 * inv;
    a.y = hi ? 0.0f     : e1 * inv;

    float best[8];
    int   bidx[8];
#pragma unroll
    for (int r = 0; r < 8; ++r) { best[r] = -3.402823466e38f; bidx[r] = 0; }

    const float2* bpv = (const float2*)bp;
#pragma unroll 5
    for (int tile = 0; tile < N_TILES; ++tile) {
        float2 bb = bpv[tile * 32 + lane];
        v2f b; b.x = bb.x; b.y = bb.y;
        v8f c = {0.f, 0.f, 0.f, 0.f, 0.f, 0.f, 0.f, 0.f};
        // D = A(16x4) x B(4x16): 16 pixels x 16 candidates, exact f32
        v8f d = __builtin_amdgcn_wmma_f32_16x16x4_f32(false, a, false, b,
                                                      (short)0, c, false, false);
        const int cand = tile * 16 + m;
#pragma unroll
        for (int r = 0; r < 8; ++r) {
            float v = d[r];
            if (v > best[r]) { best[r] = v; bidx[r] = cand; }  // strict > keeps first max
        }
    }

    // argmax reduction across the 16 lanes of each half (rows 0-7 in lanes 0-15, 8-15 in 16-31)
#pragma unroll
    for (int off = 8; off >= 1; off >>= 1) {
#pragma unroll
        for (int r = 0; r < 8; ++r) {
            float ov = __shfl_xor(best[r], off, 32);
            int   oi = __shfl_xor(bidx[r], off, 32);
            if (ov > best[r] || (ov == best[r] && oi < bidx[r])) { best[r] = ov; bidx[r] = oi; }
        }
    }

    if ((lane & 15) == 0) {
        const int mbase = hi ? 8 : 0;
#pragma unroll
        for (int r = 0; r < 8; ++r)
            out[wave * 16 + mbase + r] = (float)bidx[r];
    }
}

extern "C" void kernel_launch(void* const* d_in, const int* in_sizes, int n_in,
                              void* d_out, int out_size, void* d_ws, size_t ws_size,
                              hipStream_t stream) {
    const float* gt    = (const float*)d_in[0];   // [1,128,128]
    const float* modfs = (const float*)d_in[1];   // [10000,3]
    const float* alpha = (const float*)d_in[2];   // [3,4]
    const float* omega = (const float*)d_in[3];   // [3,4]
    const float* phi   = (const float*)d_in[4];   // [3,4]
    float* out = (float*)d_out;

    float* w      = (float*)d_ws;
    float* demod  = w;               // 30000
    float* corr   = w + 30000;       // 30000
    float* bp     = w + 60000;       // 40000 (625 tiles * 32 lanes * float2)
    float* sums   = w + 100000;      // 6: sumMod[3], sumDemod[3]
    float* consts = w + 100008;      // 3: kappas

    k_demod<<<40, 256, 0, stream>>>(alpha, omega, phi, demod);
    k_sums<<<1, 256, 0, stream>>>(modfs, demod, sums);
    k_corr<<<40, 256, 0, stream>>>(modfs, demod, sums, corr);
    k_pack<<<1, 256, 0, stream>>>(corr, sums, bp, consts);
    k_decode<<<128, 256, 0, stream>>>(gt, corr, bp, consts, out);
    (void)in_sizes; (void)n_in; (void)out_size; (void)ws_size;
}